// ViewTransformBEVPoolModel_70789650972800
// MI455X (gfx1250) — compile-verified
//
#include <hip/hip_runtime.h>
#include <cstdint>
#include <cstddef>

// Problem constants (from the reference):
//   B=1, N=6, D=118, H=32, W=88, C=80 ; grid X=Y=360, Z=1
#define NPOINTS   1993728            // 1*6*118*32*88
#define CCH       80                 // channels
#define GX        360
#define GY        360
#define SXY       (GX * GY)          // 129600 spatial cells
#define PTS_PER_CHUNK     64         // points staged per async tile
#define CHUNKS_PER_BLOCK  4
#define NCHUNKS           (NPOINTS / PTS_PER_CHUNK)            // 31152
#define NBLOCKS_SCATTER   (NCHUNKS / CHUNKS_PER_BLOCK)         // 7788
#define QUADS_PER_CHUNK   (PTS_PER_CHUNK * CCH / 4)            // 1280 float4s
#define LOADS_PER_THREAD  (QUADS_PER_CHUNK / 256)              // 5 b128 async loads

static_assert(NPOINTS % (PTS_PER_CHUNK * CHUNKS_PER_BLOCK) == 0, "chunking must be exact");
static_assert(QUADS_PER_CHUNK % 256 == 0, "async load loop must be exact");

// ---------------------------------------------------------------------------
// Zero the (C, X, Y) output accumulator (harness poisons d_out with 0xAA).
// ---------------------------------------------------------------------------
__global__ __launch_bounds__(256) void zero_out_kernel(float4* __restrict__ out, int n4) {
    int i = blockIdx.x * blockDim.x + threadIdx.x;
    if (i < n4) out[i] = make_float4(0.f, 0.f, 0.f, 0.f);
}

// ---------------------------------------------------------------------------
// Scatter-add kernel: double-buffered async global->LDS staging of feature
// tiles (CDNA5 GLOBAL_LOAD_ASYNC_TO_LDS_B128 + s_wait_asynccnt), voxel index
// computed once per point into LDS, then 80-wide atomic f32 scatter into the
// (C, X*Y) output which stays resident in the 192MB L2.
// ---------------------------------------------------------------------------
__device__ __forceinline__ void issue_chunk_async(const float* __restrict__ featsBase,
                                                  uint32_t ldsBase, int t) {
#pragma unroll
    for (int j = 0; j < LOADS_PER_THREAD; ++j) {
        const int q = t + 256 * j;                       // float4 quad index in tile
        const float* g = featsBase + (size_t)q * 4;      // 16B-aligned global source
        const uint32_t l = ldsBase + (uint32_t)(q * 16); // LDS byte offset dest
        asm volatile("global_load_async_to_lds_b128 %0, %1, off"
                     :: "v"(l), "v"(g)
                     : "memory");
    }
}

__global__ __launch_bounds__(256) void bevpool_scatter(const float* __restrict__ geom,
                                                       const float* __restrict__ feats,
                                                       float* __restrict__ out) {
    __shared__ float ldsFeat[2][PTS_PER_CHUNK * CCH];   // 2 x 20KB tiles
    __shared__ int   ldsIdx[2][PTS_PER_CHUNK];

    const int t = threadIdx.x;
    const int chunk0 = blockIdx.x * CHUNKS_PER_BLOCK;

    // Generic-pointer low 32 bits == wave-relative LDS byte offset (ISA 10.2):
    uint32_t ldsBase[2];
    ldsBase[0] = (uint32_t)(uintptr_t)&ldsFeat[0][0];
    ldsBase[1] = (uint32_t)(uintptr_t)&ldsFeat[1][0];

    // Prime the pipeline: async-load chunk 0 into buffer 0.
    issue_chunk_async(feats + (size_t)chunk0 * (PTS_PER_CHUNK * CCH), ldsBase[0], t);

    for (int i = 0; i < CHUNKS_PER_BLOCK; ++i) {
        const int buf = i & 1;
        const bool hasNext = (i + 1) < CHUNKS_PER_BLOCK;

        // Prefetch next tile into the other buffer (its previous consumer
        // finished behind the trailing barrier of iteration i-1).
        if (hasNext) {
            issue_chunk_async(feats + (size_t)(chunk0 + i + 1) * (PTS_PER_CHUNK * CCH),
                              ldsBase[buf ^ 1], t);
        }

        // One thread per point: voxel index, matching the reference f32 math.
        if (t < PTS_PER_CHUNK) {
            const size_t p = (size_t)(chunk0 + i) * PTS_PER_CHUNK + t;
            const float gx = geom[3 * p + 0];
            const float gy = geom[3 * p + 1];
            const float gz = geom[3 * p + 2];
            const float lox = -53.85f - 0.15f;   // BX - DX/2
            const float loy = -53.85f - 0.15f;
            const float loz =   0.0f - 10.0f;
            const int xi = (int)floorf((gx - lox) / 0.3f);
            const int yi = (int)floorf((gy - loy) / 0.3f);
            const int zi = (int)floorf((gz - loz) / 20.0f);
            const bool kept = (xi >= 0) & (xi < GX) & (yi >= 0) & (yi < GY) & (zi == 0);
            ldsIdx[buf][t] = kept ? (xi * GY + yi) : -1;
        }

        // Wait for THIS tile's async loads. With the next tile in flight each
        // thread has issued 5 more, so <=5 outstanding means tile i is done
        // (async loads complete in order per wave). Last tile: drain to 0.
        if (hasNext) asm volatile("s_wait_asynccnt 0x5" ::: "memory");
        else         asm volatile("s_wait_asynccnt 0x0" ::: "memory");
        __syncthreads();   // all waves drained -> LDS tile + indices visible

        // Consume: 64 pts x 20 float4 quads = 1280 quads, 5 per thread.
        // 20 consecutive threads share one point -> LDS index broadcast;
        // ds_load_b128 addresses are consecutive 16B -> conflict-free.
#pragma unroll
        for (int k = 0; k < 5; ++k) {
            const int e  = t + 256 * k;
            const int pl = e / 20;
            const int c4 = (e % 20) * 4;
            const int v  = ldsIdx[buf][pl];
            if (v >= 0) {
                const float4 f = *(const float4*)&ldsFeat[buf][pl * CCH + c4];
                float* o = out + (size_t)c4 * SXY + v;   // out[(c)*S + s], stride S per channel
                (void)__hip_atomic_fetch_add(o + 0 * (size_t)SXY, f.x, __ATOMIC_RELAXED, __HIP_MEMORY_SCOPE_AGENT);
                (void)__hip_atomic_fetch_add(o + 1 * (size_t)SXY, f.y, __ATOMIC_RELAXED, __HIP_MEMORY_SCOPE_AGENT);
                (void)__hip_atomic_fetch_add(o + 2 * (size_t)SXY, f.z, __ATOMIC_RELAXED, __HIP_MEMORY_SCOPE_AGENT);
                (void)__hip_atomic_fetch_add(o + 3 * (size_t)SXY, f.w, __ATOMIC_RELAXED, __HIP_MEMORY_SCOPE_AGENT);
            }
        }
        __syncthreads();   // buffer `buf` is refilled by the next iteration's prefetch
    }
}

// ---------------------------------------------------------------------------
// Launcher
// ---------------------------------------------------------------------------
extern "C" void kernel_launch(void* const* d_in, const int* in_sizes, int n_in,
                              void* d_out, int out_size, void* d_ws, size_t ws_size,
                              hipStream_t stream) {
    const float* geom  = (const float*)d_in[0];   // (1,6,118,32,88,3) f32
    const float* feats = (const float*)d_in[1];   // (1,6,118,32,88,80) f32
    float* out = (float*)d_out;                   // (1, 80, 360, 360) f32

    const int n4 = (CCH * SXY) / 4;               // 2,592,000 float4s
    zero_out_kernel<<<(n4 + 255) / 256, 256, 0, stream>>>((float4*)out, n4);
    bevpool_scatter<<<NBLOCKS_SCATTER, 256, 0, stream>>>(geom, feats, out);
}